// DifferentiableFDN_27032524161790
// MI455X (gfx1250) — compile-verified
//
#include <hip/hip_runtime.h>

// ---------------------------------------------------------------------------
// Differentiable FDN, collapsed to out[b,t] = sum_d w_d * s_d[b,t] with
// s_d[t] = a_d * s_d[t-1] + x[t].  Parallel-scan over time; each 16-step
// block is three 16x16x16 f32 matmuls on the WMMA pipe (12 chained
// V_WMMA_F32_16X16X4_F32 per block). All operands pre-swizzled into WMMA
// register layouts so phase C is pure b128 traffic.
// ---------------------------------------------------------------------------

#define D_ 16
#define B_ 32
#define T_ 48000
#define P_ 750          // time partitions
#define LP_ 64          // samples per partition (4 WMMA blocks of 16)
#define BLKS_PER_P 4

typedef __attribute__((ext_vector_type(2))) float v2f;
typedef __attribute__((ext_vector_type(4))) float v4f;
typedef __attribute__((ext_vector_type(8))) float v8f;

// workspace layout (in floats)
#define WS_A      0                          // decays a_d            (16)
#define WS_W      16                         // collapsed weights w_d (16)
#define WS_WL     64                         // per-lane w  [lane][v]  (256)
#define WS_A16L   320                        // per-lane a^16 [lane][v](256)
#define WS_TAB    576                        // 12 A-matrix chunks * 64 = 768
#define WS_XS     2048                       // x swizzled, B-layout  = 1,536,000
#define WS_E      (WS_XS + T_*B_)            // partition end states  = 384,000
#define WS_CARRYS (WS_E + P_*B_*D_)          // carries, B-layout     = 384,000
// total ~9.23 MB of d_ws

__device__ __forceinline__ float sigmoidf_(float z) { return 1.0f / (1.0f + expf(-z)); }
__device__ __forceinline__ float powi_(float a, int n) {
    float r = 1.0f;
    for (int i = 0; i < n; ++i) r *= a;
    return r;
}

// lane/reg -> delay index for the B-matrix / state layout:
//   reg v, lane half h  ->  d = 4*(v>>1) + 2h + (v&1)
// ---------------------------------------------------------------------------
// Phase 0: parameters + constant WMMA A-matrix tables + per-lane param tables.
//   chunks 0-3 : Lsum : Lsum[k,j] = sum_d w_d a_d^(k-j), j<=k
//   chunks 4-7 : Pmat : Pmat[k,d] = a_d^(k+1)
//   chunks 8-11: Rmat : Rmat[m,j] = a_{sigma(m)}^(15-j)  (rows permuted so the
//                WMMA C-layout result lands directly in the state layout)
// A-matrix VGPR layout per chunk: lane L, reg ra -> M = L&15, K = ra + 2*(L>>4)
// ---------------------------------------------------------------------------
__global__ void fdn_setup(const float* __restrict__ lk, const float* __restrict__ ar,
                          const float* __restrict__ br, const float* __restrict__ H,
                          float* __restrict__ ws) {
    const int t = threadIdx.x;
    if (t < D_) {
        const float max_delay = 50.0f * 16000.0f / 1000.0f;   // 800 samples
        float kappa = 1.0f + sigmoidf_(lk[t]) * (max_delay - 1.0f);
        float a = expf(-1.0f / kappa);
        a = fminf(fmaxf(a, 0.0f), 0.9999f);
        ws[WS_A + t] = a;
        float s = 0.0f;
        for (int i = 0; i < D_; ++i) s += sigmoidf_(ar[i]) * H[i * D_ + t];
        ws[WS_W + t] = (sigmoidf_(br[t]) + s) * (1.0f / (float)D_);
    }
    __syncthreads();

    // per-lane parameter tables, [lane][v] so phase C loads them as v8f
    for (int e = t; e < 2 * 256; e += blockDim.x) {
        const int which = e >> 8, idx = e & 255;
        const int lane = idx >> 3, v = idx & 7;
        const int h = lane >> 4;
        const int d = 4 * (v >> 1) + 2 * h + (v & 1);
        const float a = ws[WS_A + d];
        if (which == 0) ws[WS_WL + idx] = ws[WS_W + d];
        else            ws[WS_A16L + idx] = powi_(a, 16);
    }

    // WMMA A-matrix tables
    for (int e = t; e < 12 * 64; e += blockDim.x) {
        const int chunk = e >> 6, pos = e & 63;
        const int lane = pos >> 1, ra = pos & 1;
        const int h = lane >> 4, m = lane & 15;
        const int kap = ra + 2 * h;                 // K index within chunk
        float val = 0.0f;
        if (chunk < 4) {                             // Lsum
            const int j = 4 * chunk + kap;
            if (j <= m)
                for (int d = 0; d < D_; ++d)
                    val += ws[WS_W + d] * powi_(ws[WS_A + d], m - j);
        } else if (chunk < 8) {                      // Pmat
            const int d = 4 * (chunk - 4) + kap;
            val = powi_(ws[WS_A + d], m + 1);
        } else {                                     // Rmat (sigma-permuted rows)
            int sg;
            if (m < 8) sg = 4 * (m >> 1) + (m & 1);
            else { const int v = m - 8; sg = 4 * (v >> 1) + 2 + (v & 1); }
            const int j = 4 * (chunk - 8) + kap;
            val = powi_(ws[WS_A + sg], 15 - j);
        }
        ws[WS_TAB + e] = val;
    }
}

// ---------------------------------------------------------------------------
// Swizzle x (B,T) directly into the per-block WMMA B-matrix layout:
//   xS[((tblk*2 + g)*32 + h*16 + bl)*8 + v],  b = g*16+bl, t = tblk*16 + j,
//   j = 4*(v>>1) + 2h + (v&1).  Coalesced reads; writes land in L2.
// ---------------------------------------------------------------------------
__global__ void fdn_swizzle(const float* __restrict__ x, float* __restrict__ ws) {
    const int id = blockIdx.x * blockDim.x + threadIdx.x;
    if (id >= B_ * T_) return;
    const int b = id / T_, t = id % T_;          // consecutive threads: consecutive t
    const float val = x[id];
    const int tblk = t >> 4, j = t & 15;
    const int g = b >> 4, bl = b & 15;
    const int kk = j >> 2, kap = j & 3;
    const int h = kap >> 1, r = kap & 1;
    const int v = 2 * kk + r;
    ws[WS_XS + ((((tblk << 1) | g) * 32 + (h << 4) + bl) << 3) + v] = val;
}

// ---------------------------------------------------------------------------
// Phase A: zero-init local end-state per (partition, b, d):
//   E = sum_i a^(LP-1-i) x[t0+i]   (serial 64-FMA chain, 384k threads)
// lanes sharing (b,p) differ only in d -> broadcast loads from x.
// ---------------------------------------------------------------------------
__global__ void fdn_partial(const float* __restrict__ x, float* __restrict__ ws) {
    const int tid = blockIdx.x * blockDim.x + threadIdx.x;
    if (tid >= P_ * B_ * D_) return;
    const int p = tid >> 9;
    const int r = tid & 511;
    const int b = r >> 4, d = r & 15;
    const float a = ws[WS_A + d];
    const float* xp = x + b * T_ + p * LP_;
    float acc = 0.0f;
#pragma unroll 4
    for (int i = 0; i < LP_; ++i) acc = fmaf(a, acc, xp[i]);
    ws[WS_E + (p * B_ + b) * D_ + d] = acc;
}

// ---------------------------------------------------------------------------
// Phase B: cross-partition carry scan (512 threads x 750 sequential steps).
// carry[p] = state entering partition p, stored directly in the phase-C
// state layout: carryS[((p*2+g)*32 + h*16 + bl)*8 + v].
// ---------------------------------------------------------------------------
__global__ void fdn_carry(float* __restrict__ ws) {
    const int tid = blockIdx.x * blockDim.x + threadIdx.x;
    if (tid >= B_ * D_) return;
    const int b = tid >> 4, d = tid & 15;
    const int g = b >> 4, bl = b & 15;
    const int h = (d >> 1) & 1, v = 2 * (d >> 2) + (d & 1);
    const int laneSlot = (((h << 4) + bl) << 3) + v;
    const float a = ws[WS_A + d];
    float aL = a;
    for (int q = 0; q < 6; ++q) aL *= aL;      // a^64
    float c = 0.0f;
    for (int p = 0; p < P_; ++p) {
        ws[WS_CARRYS + ((((p << 1) | g) * 32) << 3) + laneSlot] = c;
        c = fmaf(aL, c, ws[WS_E + (p * B_ + b) * D_ + d]);
    }
}

// ---------------------------------------------------------------------------
// Phase C: one wave per (partition, batch-group of 16). Per 16-step block:
//   acc = Pmat @ (w*state)  +  Lsum @ X     (carry correction + local output)
//   rr  = Rmat @ X                          (block-end rows, state layout)
//   state = a^16 * state + rr
// All operands are v8f (2x b128) loads; outputs 2x b128 stores per block.
// ---------------------------------------------------------------------------
__global__ void __launch_bounds__(256) fdn_main(const float* __restrict__ ws,
                                                float* __restrict__ out) {
    const int lane = threadIdx.x & 31;
    const int widx = blockIdx.x * (blockDim.x >> 5) + (threadIdx.x >> 5);
    if (widx >= 2 * P_) return;                // uniform per-wave exit
    const int p = widx >> 1, g = widx & 1;
    const int h = lane >> 4, bl = lane & 15;

    // per-lane parameter vectors (state layout)
    const v8f wlv   = *(const v8f*)(ws + WS_WL   + lane * 8);
    const v8f a16lv = *(const v8f*)(ws + WS_A16L + lane * 8);

    // constant A matrices (v2f per chunk per lane)
    v2f Lm[4], Pm[4], Rm[4];
#pragma unroll
    for (int kk = 0; kk < 4; ++kk) {
        Lm[kk] = *(const v2f*)(ws + WS_TAB + (kk)     * 64 + lane * 2);
        Pm[kk] = *(const v2f*)(ws + WS_TAB + (kk + 4) * 64 + lane * 2);
        Rm[kk] = *(const v2f*)(ws + WS_TAB + (kk + 8) * 64 + lane * 2);
    }

    // running state (pre-swizzled by phase B)
    v8f stv = *(const v8f*)(ws + WS_CARRYS + (widx * 32 + lane) * 8);

    const float* xS = ws + WS_XS;
    int tb = p * BLKS_PER_P;                   // 16-sample block index
    int t0 = p * LP_;
    for (int blk = 0; blk < BLKS_PER_P; ++blk) {
        // X operand: one v8f per lane, already in B-matrix layout
        const v8f xv = *(const v8f*)(xS + (((tb << 1) | g) * 32 + lane) * 8);
        const v2f* xc = (const v2f*)&xv;

        // Q = w * state (element-wise), sliced per K-chunk
        v8f qv = wlv * stv;
        const v2f* qc = (const v2f*)&qv;

        v8f acc = {0.f, 0.f, 0.f, 0.f, 0.f, 0.f, 0.f, 0.f};
#pragma unroll
        for (int kk = 0; kk < 4; ++kk)
            acc = __builtin_amdgcn_wmma_f32_16x16x4_f32(false, Pm[kk], false, qc[kk],
                                                        (short)0, acc, false, false);
        v8f rr = {0.f, 0.f, 0.f, 0.f, 0.f, 0.f, 0.f, 0.f};
#pragma unroll
        for (int kk = 0; kk < 4; ++kk) {
            acc = __builtin_amdgcn_wmma_f32_16x16x4_f32(false, Lm[kk], false, xc[kk],
                                                        (short)0, acc, false, false);
            rr  = __builtin_amdgcn_wmma_f32_16x16x4_f32(false, Rm[kk], false, xc[kk],
                                                        (short)0, rr, false, false);
        }
        stv = a16lv * stv + rr;                // pk_fma state update

        // store: C-layout reg r -> t = t0 + 8h + r (8 consecutive samples/lane)
        float* op = out + (g * 16 + bl) * T_ + t0 + 8 * h;
        const v4f* ap = (const v4f*)&acc;
        *(v4f*)op       = ap[0];
        *(v4f*)(op + 4) = ap[1];

        ++tb;
        t0 += 16;
    }
}

// ---------------------------------------------------------------------------
extern "C" void kernel_launch(void* const* d_in, const int* in_sizes, int n_in,
                              void* d_out, int out_size, void* d_ws, size_t ws_size,
                              hipStream_t stream) {
    const float* x  = (const float*)d_in[0];
    const float* lk = (const float*)d_in[1];
    const float* ar = (const float*)d_in[2];
    const float* br = (const float*)d_in[3];
    const float* H  = (const float*)d_in[4];
    float* ws  = (float*)d_ws;     // uses ~9.3 MB
    float* out = (float*)d_out;

    hipLaunchKernelGGL(fdn_setup,   dim3(1),                          dim3(256), 0, stream, lk, ar, br, H, ws);
    hipLaunchKernelGGL(fdn_swizzle, dim3((B_ * T_ + 255) / 256),      dim3(256), 0, stream, x, ws);
    hipLaunchKernelGGL(fdn_partial, dim3((P_ * B_ * D_ + 255) / 256), dim3(256), 0, stream, x, ws);
    hipLaunchKernelGGL(fdn_carry,   dim3(2),                          dim3(256), 0, stream, ws);
    hipLaunchKernelGGL(fdn_main,    dim3((2 * P_ + 7) / 8),           dim3(256), 0, stream, ws, out);
}